// Convolve_77644418777227
// MI455X (gfx1250) — compile-verified
//
#include <hip/hip_runtime.h>

// Per-pixel predicted-filter 5x5x3 convolution (KPN). Memory-bound streaming
// kernel: image tile staged in LDS via CDNA5 async global->LDS DMA, filts
// streamed with non-temporal hints, VALU FMA reduction.
//
// Roofline: 331 MB total traffic (filts 314.6 MB read-once dominates) at
// 23.3 TB/s => ~14 us floor; 157 MFLOP => ~11 TFLOPS => memory-bound >100x.
// WMMA inapplicable (independent per-pixel dots => only the diagonal of a
// 16x16 WMMA result is useful => 16x wasted f32 FLOPs); the CDNA5 win is
// global_load_async_to_lds + s_wait_asynccnt + NT cache hints.

#define KSZ   5
#define PAD   2
#define TS    16                    // output tile edge per block
#define TROWS (TS + 2 * PAD)        // 20 image rows per tile
#define ROWF  (TROWS * 3)           // 60 floats per staged row
#define TILE_FLOATS (TROWS * ROWF)  // 1200 floats = 4800 B LDS
#define IMG_H 512
#define IMG_W 512
#define IMG_ROWF (IMG_W * 3)        // 1536 floats per image row
#define FLEN  (KSZ * KSZ * 3)       // 75 filter taps per pixel

#if defined(__HIP_DEVICE_COMPILE__) && __has_builtin(__builtin_amdgcn_global_load_async_to_lds_b32)
#define USE_ASYNC_LDS 1
#else
#define USE_ASYNC_LDS 0
#endif

__global__ __launch_bounds__(256)
void kpn_conv5_kernel(const float* __restrict__ img,
                      const float* __restrict__ filts,
                      float* __restrict__ out) {
    __shared__ float tile[TILE_FLOATS];

    const int tid = threadIdx.x;
    const int bx  = blockIdx.x & 31;          // tile x index (512/16 = 32)
    const int by  = (blockIdx.x >> 5) & 31;   // tile y index
    const int b   = blockIdx.x >> 10;         // batch
    const int x0  = bx * TS;
    const int y0  = by * TS;
    const long imgBase = (long)b * (IMG_H * IMG_ROWF);

#if USE_ASYNC_LDS
    // Phase 1: zero the tile (halo slots outside the image keep 0 == jnp.pad).
    for (int i = tid; i < TILE_FLOATS; i += 256) tile[i] = 0.0f;
    __syncthreads();  // all ds stores visible before async writes land

    // Phase 2: async DMA of in-bounds texels straight into LDS (no VGPR trip).
    typedef __attribute__((address_space(1))) int gm_i32_t;
    typedef __attribute__((address_space(3))) int lds_i32_t;
    for (int i = tid; i < TILE_FLOATS; i += 256) {
        const int r  = i / ROWF;               // staged row 0..19
        const int e  = i - r * ROWF;           // float within row 0..59
        const int gy = y0 - PAD + r;           // image row
        const int ge = (x0 - PAD) * 3 + e;     // float within image row
        if ((unsigned)gy < (unsigned)IMG_H && (unsigned)ge < (unsigned)IMG_ROWF) {
            const float* src = img + imgBase + (long)gy * IMG_ROWF + ge;
            __builtin_amdgcn_global_load_async_to_lds_b32(
                (gm_i32_t*)src, (lds_i32_t*)&tile[i], 0, 0);
        }
    }
#if __has_builtin(__builtin_amdgcn_s_wait_asynccnt)
    __builtin_amdgcn_s_wait_asynccnt(0);
#else
    asm volatile("s_wait_asynccnt 0" ::: "memory");
#endif
    __syncthreads();  // all waves' async tile writes complete
#else
    // Fallback staging: ordinary load + ds_store.
    for (int i = tid; i < TILE_FLOATS; i += 256) {
        const int r  = i / ROWF;
        const int e  = i - r * ROWF;
        const int gy = y0 - PAD + r;
        const int ge = (x0 - PAD) * 3 + e;
        float v = 0.0f;
        if ((unsigned)gy < (unsigned)IMG_H && (unsigned)ge < (unsigned)IMG_ROWF)
            v = img[imgBase + (long)gy * IMG_ROWF + ge];
        tile[i] = v;
    }
    __syncthreads();
#endif

    const int tx  = tid & 15;
    const int ty  = tid >> 4;
    const int x   = x0 + tx;
    const int y   = y0 + ty;
    const int pix = ((b * IMG_H) + y) * IMG_W + x;
    const float* __restrict__ f = filts + (long)pix * FLEN;

    // Load phase: issue all 75 non-temporal filter loads up front so the
    // whole 300 B/pixel stream is in flight (max MLP; LOADcnt throttles at
    // 63 outstanding per wave). NT keeps the read-once 315 MB stream from
    // evicting reusable image tiles out of the 192 MB L2.
    float fv[FLEN];
#pragma unroll
    for (int k = 0; k < FLEN; ++k)
        fv[k] = __builtin_nontemporal_load(f + k);

    // FMA phase: 25 taps x 3 channels against the LDS tile.
    float acc = 0.0f;
#pragma unroll
    for (int t = 0; t < KSZ * KSZ; ++t) {
        const int dy = t / KSZ;
        const int dx = t - dy * KSZ;
        const float* ls = &tile[(ty + dy) * ROWF + (tx + dx) * 3];
        acc = fmaf(fv[3 * t + 0], ls[0], acc);
        acc = fmaf(fv[3 * t + 1], ls[1], acc);
        acc = fmaf(fv[3 * t + 2], ls[2], acc);
    }
    __builtin_nontemporal_store(acc, out + pix);
}

extern "C" void kernel_launch(void* const* d_in, const int* in_sizes, int n_in,
                              void* d_out, int out_size, void* d_ws, size_t ws_size,
                              hipStream_t stream) {
    (void)in_sizes; (void)n_in; (void)d_ws; (void)ws_size; (void)out_size;
    const float* img   = (const float*)d_in[0];   // [4,512,512,3] f32
    const float* filts = (const float*)d_in[1];   // [4,512,512,75] f32
    float* out         = (float*)d_out;           // [4,512,512] f32

    const int grid = 4 * 32 * 32;  // B * (H/TS) * (W/TS)
    kpn_conv5_kernel<<<grid, 256, 0, stream>>>(img, filts, out);
}